// DiscMaker_10642928960163
// MI455X (gfx1250) — compile-verified
//
#include <hip/hip_runtime.h>
#include <math.h>

// ---------------------------------------------------------------------------
// Recurrent mixture-of-Kalman GRU scan for MI455X (gfx1250), round 2.
// B=128 batch rows -> 8 workgroups x 16 rows; 16 rows == WMMA M dimension.
// fp32 WMMA 16x16x4 (exact reference precision; workload is dependency-
// latency bound at ~49 GFLOP total, so fp32 matrix rate is not the limiter).
// W_h / W_x B-fragments are hoisted into per-wave registers (t-invariant),
// activations use the TRANS pipe (v_exp_f32) instead of branchy libm tanh.
// ---------------------------------------------------------------------------

typedef __attribute__((ext_vector_type(2))) float v2f;
typedef __attribute__((ext_vector_type(8))) float v8f;

// A fragment (16x4 f32): lanes 0-15 rows M=0..15 hold K={k0,k0+1},
// lanes 16-31 hold K={k0+2,k0+3}. (ISA 7.12.2 "32-bit A-Matrix 16x4")
__device__ __forceinline__ v2f frag_a(const float* A, int ld, int k0) {
  int lane = threadIdx.x & 31;
  int row  = lane & 15;
  int kk   = k0 + ((lane >> 4) << 1);
  v2f a;
  a.x = A[row * ld + kk];
  a.y = A[row * ld + kk + 1];
  return a;
}

// B fragment (4x16 f32): lanes hold N columns, VGPRs hold K rows (mirrors A).
__device__ __forceinline__ v2f frag_b(const float* W, int ld, int k0, int c0) {
  int lane = threadIdx.x & 31;
  int col  = c0 + (lane & 15);
  int kk   = k0 + ((lane >> 4) << 1);
  v2f b;
  b.x = W[kk * ld + col];
  b.y = W[(kk + 1) * ld + col];
  return b;
}

// B fragment with K-range guard (for the 65-row W_x tail).
__device__ __forceinline__ v2f frag_b_kguard(const float* W, int ld, int k0,
                                             int c0, int kmax) {
  int lane = threadIdx.x & 31;
  int col  = c0 + (lane & 15);
  int kk   = k0 + ((lane >> 4) << 1);
  v2f b;
  b.x = (kk     < kmax) ? W[kk * ld + col]       : 0.0f;
  b.y = (kk + 1 < kmax) ? W[(kk + 1) * ld + col] : 0.0f;
  return b;
}

__device__ __forceinline__ v8f wmma4(v2f a, v2f b, v8f c) {
  return __builtin_amdgcn_wmma_f32_16x16x4_f32(
      /*neg_a=*/false, a, /*neg_b=*/false, b,
      /*c_mod=*/(short)0, c, /*reuse_a=*/false, /*reuse_b=*/false);
}

// tanh via the TRANS pipe: tanh(x) = 1 - 2/(exp(2x)+1). Branch-free,
// saturates correctly at +/-inf. Avoids libm's EXEC-divergent polynomial.
__device__ __forceinline__ float fast_tanh(float x) {
  float e = __expf(2.0f * x);
  return 1.0f - __fdividef(2.0f, e + 1.0f);
}
__device__ __forceinline__ float fast_sigmoid(float x) {
  return __fdividef(1.0f, 1.0f + __expf(-x));
}

__global__ __launch_bounds__(256) void kalman_gru_scan(
    const float* __restrict__ x,      // (B,T,32)
    const float* __restrict__ y,      // (B,T)
    const float* __restrict__ W_enc,  // (32,64)
    const float* __restrict__ b_enc,  // (64)
    const float* __restrict__ W_in,   // (8,32,32)
    const float* __restrict__ W_rec,  // (8,32,32)
    const float* __restrict__ b_s,    // (8,32)
    const float* __restrict__ W_x,    // (65,384)
    const float* __restrict__ W_h,    // (128,384)
    const float* __restrict__ b_x,    // (384)
    const float* __restrict__ b_h,    // (384)
    const float* __restrict__ W_gate, // (128,8)
    const float* __restrict__ b_gate, // (8)
    float* __restrict__ out)          // (B,T)
{
  constexpr int T = 2048, IN_ = 32, S = 32, M = 8, E = 64, H = 128;
  constexpr int CIN = 65, G = 384;

  __shared__ float sh_x[16 * 32];       // x_t tile
  __shared__ float sh_inp[16 * 68];     // [encoded(64) | err(1) | zero pad(3)]
  __shared__ float sh_h[16 * 128];      // GRU hidden carry
  __shared__ float sh_ns[16 * 8 * 32];  // new_state (b, m, s)
  __shared__ float sh_kst[16 * 32];     // kstate carry
  __shared__ float sh_gate[16 * 8];     // softmax gate
  __shared__ float sh_sc0[16 * 16];     // gate logits partial (K 0..63)
  __shared__ float sh_sc1[16 * 16];     // gate logits partial (K 64..127)
  __shared__ float sh_y[16];
  __shared__ float sh_err[16];

  const int tid  = threadIdx.x;
  const int wave = tid >> 5;
  const int lane = tid & 31;
  const int b0   = blockIdx.x * 16;

  // ---- per-lane loop-invariant constants --------------------------------
  const int colb = lane & 15;
  const int rb   = (lane >> 4) << 3;   // C/D row base: 0 or 8
  const int zc0  = wave * 16;          // this wave's h-column tile base
  const int hc   = zc0 + colb;         // this lane's h column (0..127)

  const float bias_z  = b_x[hc]       + b_h[hc];
  const float bias_r  = b_x[128 + hc] + b_h[128 + hc];
  const float bias_xn = b_x[256 + hc];
  const float bias_hn = b_h[256 + hc];
  const float bias_s0 = b_s[wave * 32 + colb];
  const float bias_s1 = b_s[wave * 32 + 16 + colb];
  const float bias_e  = (wave < 4) ? b_enc[wave * 16 + colb] : 0.0f;

  // ---- hoist t-invariant W_h / W_x B-fragments into registers -----------
  // W_h: 32 K-slices x 3 gate tiles = 192 VGPRs/lane.
  v2f whz[32], whr[32], whn[32];
  #pragma unroll
  for (int i = 0; i < 32; ++i) {
    whz[i] = frag_b(W_h, G, 4 * i, zc0);
    whr[i] = frag_b(W_h, G, 4 * i, 128 + zc0);
    whn[i] = frag_b(W_h, G, 4 * i, 256 + zc0);
  }
  // W_x: 17 K-slices (last one K-guarded at 65 rows) x 3 tiles = 102 VGPRs.
  v2f wxz[17], wxr[17], wxn[17];
  #pragma unroll
  for (int i = 0; i < 16; ++i) {
    wxz[i] = frag_b(W_x, G, 4 * i, zc0);
    wxr[i] = frag_b(W_x, G, 4 * i, 128 + zc0);
    wxn[i] = frag_b(W_x, G, 4 * i, 256 + zc0);
  }
  wxz[16] = frag_b_kguard(W_x, G, 64, zc0,       CIN);
  wxr[16] = frag_b_kguard(W_x, G, 64, 128 + zc0, CIN);
  wxn[16] = frag_b_kguard(W_x, G, 64, 256 + zc0, CIN);

  // ---- zero-init carries and inp padding --------------------------------
  for (int i = tid; i < 16 * H; i += 256) sh_h[i] = 0.0f;
  for (int i = tid; i < 16 * S; i += 256) sh_kst[i] = 0.0f;
  for (int i = tid; i < 16 * 68; i += 256) sh_inp[i] = 0.0f;
  if (tid < 16) sh_err[tid] = 0.0f;
  __syncthreads();

  for (int t = 0; t < T; ++t) {
    // ---- stage 0: load x_t tile, y_t, inject err into inp col 64 --------
    for (int i = tid; i < 16 * IN_; i += 256) {
      int row = i >> 5, col = i & 31;
      sh_x[i] = x[(size_t)(b0 + row) * T * IN_ + (size_t)t * IN_ + col];
    }
    if (tid < 16) {
      sh_y[tid] = y[(size_t)(b0 + tid) * T + t];
      sh_inp[tid * 68 + 64] = sh_err[tid];
      if (t + 1 < T)  // gfx1250 global_prefetch_b8: next step's x rows
        __builtin_prefetch(&x[(size_t)(b0 + tid) * T * IN_ +
                              (size_t)(t + 1) * IN_], 0, 1);
    }
    __syncthreads();

    // ---- stage 1a: new_state[m=wave] = tanh(x@W_in[m]+k@W_rec[m]+b_s) ---
    {
      const int m = wave;
      const float* Wi = W_in  + m * IN_ * S;   // small: stream from L2
      const float* Wr = W_rec + m * S * S;
      v8f c0 = {}, c1 = {};
      #pragma unroll
      for (int k0 = 0; k0 < IN_; k0 += 4) {
        v2f a = frag_a(sh_x, 32, k0);
        c0 = wmma4(a, frag_b(Wi, S, k0, 0),  c0);
        c1 = wmma4(a, frag_b(Wi, S, k0, 16), c1);
      }
      #pragma unroll
      for (int k0 = 0; k0 < S; k0 += 4) {
        v2f a = frag_a(sh_kst, 32, k0);
        c0 = wmma4(a, frag_b(Wr, S, k0, 0),  c0);
        c1 = wmma4(a, frag_b(Wr, S, k0, 16), c1);
      }
      #pragma unroll
      for (int v = 0; v < 8; ++v) {
        int row = rb + v;
        sh_ns[row * 256 + m * 32 + colb]      = fast_tanh(c0[v] + bias_s0);
        sh_ns[row * 256 + m * 32 + 16 + colb] = fast_tanh(c1[v] + bias_s1);
      }
    }
    // ---- stage 1b: encoded = tanh(x@W_enc + b_enc) (waves 0-3) ----------
    if (wave < 4) {
      int c0c = wave * 16;
      v8f c = {};
      #pragma unroll
      for (int k0 = 0; k0 < IN_; k0 += 4)
        c = wmma4(frag_a(sh_x, 32, k0), frag_b(W_enc, E, k0, c0c), c);
      #pragma unroll
      for (int v = 0; v < 8; ++v)
        sh_inp[(rb + v) * 68 + c0c + colb] = fast_tanh(c[v] + bias_e);
    }
    __syncthreads();

    // ---- stage 2: fused GRU GEMMs from register-resident weights --------
    // z,r fuse x-path and h-path; n keeps them split (n = tanh(xn + r*hn)).
    float hnew[8];
    {
      v8f Cz = {}, Cr = {}, Cxn = {}, Chn = {};
      #pragma unroll
      for (int i = 0; i < 16; ++i) {             // inp @ W_x, K=0..63
        v2f a = frag_a(sh_inp, 68, 4 * i);
        Cz  = wmma4(a, wxz[i], Cz);
        Cr  = wmma4(a, wxr[i], Cr);
        Cxn = wmma4(a, wxn[i], Cxn);
      }
      { // K tail 64..67: inp cols 65..67 zero-padded, W_x frag pre-guarded
        v2f a = frag_a(sh_inp, 68, 64);
        Cz  = wmma4(a, wxz[16], Cz);
        Cr  = wmma4(a, wxr[16], Cr);
        Cxn = wmma4(a, wxn[16], Cxn);
      }
      #pragma unroll
      for (int i = 0; i < 32; ++i) {             // h @ W_h, K=128
        v2f a = frag_a(sh_h, H, 4 * i);
        Cz  = wmma4(a, whz[i], Cz);
        Cr  = wmma4(a, whr[i], Cr);
        Chn = wmma4(a, whn[i], Chn);
      }
      #pragma unroll
      for (int v = 0; v < 8; ++v) {
        int row = rb + v;
        float z = fast_sigmoid(Cz[v] + bias_z);
        float r = fast_sigmoid(Cr[v] + bias_r);
        float n = fast_tanh((Cxn[v] + bias_xn) + r * (Chn[v] + bias_hn));
        hnew[v] = (1.0f - z) * n + z * sh_h[row * H + hc];
      }
    }
    __syncthreads();  // all waves finished reading old sh_h
    #pragma unroll
    for (int v = 0; v < 8; ++v) sh_h[(rb + v) * H + hc] = hnew[v];
    __syncthreads();

    // ---- stage 3: gate logits = h_new @ W_gate, K split over waves 0/1 --
    if (wave < 2) {
      float* dst = (wave == 0) ? sh_sc0 : sh_sc1;
      const int kbase = wave * 64;
      v8f c = {};
      #pragma unroll
      for (int i = 0; i < 16; ++i) {
        int k0 = kbase + 4 * i;
        v2f a = frag_a(sh_h, H, k0);
        int kk = k0 + ((lane >> 4) << 1);
        v2f b;
        b.x = (colb < M) ? W_gate[kk * M + colb]       : 0.0f;
        b.y = (colb < M) ? W_gate[(kk + 1) * M + colb] : 0.0f;
        c = wmma4(a, b, c);
      }
      if (colb < M) {
        #pragma unroll
        for (int v = 0; v < 8; ++v) dst[(rb + v) * 16 + colb] = c[v];
      }
    }
    __syncthreads();

    // ---- stage 4: row-wise softmax over M=8 (16 threads) ----------------
    if (tid < 16) {
      float s[M], mx = -1e30f;
      #pragma unroll
      for (int m = 0; m < M; ++m) {
        s[m] = sh_sc0[tid * 16 + m] + sh_sc1[tid * 16 + m] + b_gate[m];
        mx = fmaxf(mx, s[m]);
      }
      float sum = 0.0f;
      #pragma unroll
      for (int m = 0; m < M; ++m) { s[m] = __expf(s[m] - mx); sum += s[m]; }
      float inv = __fdividef(1.0f, sum);
      #pragma unroll
      for (int m = 0; m < M; ++m) sh_gate[tid * M + m] = s[m] * inv;
    }
    __syncthreads();

    // ---- stage 5: kstate = sum_m gate*new_state; pred; err --------------
    for (int i = tid; i < 16 * S; i += 256) {
      int row = i >> 5, col = i & 31;
      float acc = 0.0f;
      #pragma unroll
      for (int m = 0; m < M; ++m)
        acc += sh_gate[row * M + m] * sh_ns[row * 256 + m * 32 + col];
      sh_kst[i] = acc;
      if (col == S - 1) {
        out[(size_t)(b0 + row) * T + t] = acc;   // pred = kstate[:, -1]
        sh_err[row] = acc - sh_y[row];
      }
    }
    __syncthreads();  // carries stable before next step
  }
}

extern "C" void kernel_launch(void* const* d_in, const int* in_sizes, int n_in,
                              void* d_out, int out_size, void* d_ws, size_t ws_size,
                              hipStream_t stream) {
  (void)in_sizes; (void)n_in; (void)out_size; (void)d_ws; (void)ws_size;
  const float* x      = (const float*)d_in[0];
  const float* y      = (const float*)d_in[1];
  const float* W_enc  = (const float*)d_in[2];
  const float* b_enc  = (const float*)d_in[3];
  const float* W_in   = (const float*)d_in[4];
  const float* W_rec  = (const float*)d_in[5];
  const float* b_s    = (const float*)d_in[6];
  const float* W_x    = (const float*)d_in[7];
  const float* W_h    = (const float*)d_in[8];
  const float* b_x    = (const float*)d_in[9];
  const float* b_h    = (const float*)d_in[10];
  const float* W_gate = (const float*)d_in[11];
  const float* b_gate = (const float*)d_in[12];
  float* out = (float*)d_out;

  // B=128 rows -> 8 workgroups of 16 rows (WMMA M dim); 256 threads = 8 waves.
  kalman_gru_scan<<<dim3(8), dim3(256), 0, stream>>>(
      x, y, W_enc, b_enc, W_in, W_rec, b_s, W_x, W_h, b_x, b_h,
      W_gate, b_gate, out);
}